// RACNN_86431921865104
// MI455X (gfx1250) — compile-verified
//
#include <hip/hip_runtime.h>
#include <hip/hip_bf16.h>
#include <math.h>

// Problem constants
#define SDIM  448
#define OUTN  224
#define BATCH 64
#define CH    3
#define NT    14      // 224/16 output tiles per axis
#define KW    48      // band window (max band span per 16-out tile is ~34)
#define KCH   12      // KW / 4 (K-chunks for V_WMMA_F32_16X16X4_F32)
#define IST   52      // imgT column stride (dwords): bank-conflict-free for b64 reads
#define TST   52      // tmpT row stride (dwords)

typedef float v2f __attribute__((ext_vector_type(2)));
typedef float v8f __attribute__((ext_vector_type(8)));

__device__ __forceinline__ float sigm(float x) { return 1.0f / (1.0f + expf(-x)); }
__device__ __forceinline__ int   imin(int a, int b) { return a < b ? a : b; }

// ---------------------------------------------------------------------------
// Kernel 1: per-batch crop parameters, interp indices/fractions, masks, band
// starts. One block per batch, 224 threads.
// ---------------------------------------------------------------------------
__global__ __launch_bounds__(OUTN) void racnn_prep(
    const float* __restrict__ locs,
    float* __restrict__ rowFr, float* __restrict__ colFr,
    float* __restrict__ mrow,  float* __restrict__ mcol,
    int*   __restrict__ rowI0, int*   __restrict__ colI0,
    int*   __restrict__ rowKb, int*   __restrict__ colKb)
{
    int b = blockIdx.x;
    int t = threadIdx.x;
    const float fS = 448.0f;

    float tl = fminf(fmaxf(locs[b*3 + 2], fS / 3.0f), fS * 2.0f / 3.0f);
    float tx = fminf(fmaxf(locs[b*3 + 0], tl), fS - tl);
    float ty = fminf(fmaxf(locs[b*3 + 1], tl), fS - tl);

    float w_off = fmaxf(floorf(tx - tl), 0.0f);
    float w_end = (tx + tl < fS) ? floorf(tx + tl) : fS;
    float h_off = fmaxf(floorf(ty - tl), 0.0f);
    float h_end = (ty + tl < fS) ? floorf(ty + tl) : fS;

    float sw = (w_end - w_off - 1.0f) / (float)(OUTN - 1);
    float sh = (h_end - h_off - 1.0f) / (float)(OUTN - 1);

    // interp source index + fraction for this output coordinate
    {
        float src = w_off + (float)t * sw;
        int   i0  = (int)fminf(fmaxf(floorf(src), 0.0f), (float)(SDIM - 1));
        rowI0[b*OUTN + t] = i0;
        rowFr[b*OUTN + t] = src - (float)i0;

        float src2 = h_off + (float)t * sh;
        int   j0   = (int)fminf(fmaxf(floorf(src2), 0.0f), (float)(SDIM - 1));
        colI0[b*OUTN + t] = j0;
        colFr[b*OUTN + t] = src2 - (float)j0;
    }

    // sigmoid edge masks over the full 448 axis
    for (int i = t; i < SDIM; i += OUTN) {
        float fi = (float)i;
        mrow[b*SDIM + i] = sigm(10.0f * (fi - w_off)) - sigm(10.0f * (fi - w_end));
        mcol[b*SDIM + i] = sigm(10.0f * (fi - h_off)) - sigm(10.0f * (fi - h_end));
    }

    // band start per 16-output tile (clamped so the 48-window stays in bounds)
    if (t < NT) {
        float src = w_off + (float)(16 * t) * sw;
        int   i0  = (int)fminf(fmaxf(floorf(src), 0.0f), (float)(SDIM - 1));
        rowKb[b*NT + t] = imin(i0, SDIM - KW);

        float src2 = h_off + (float)(16 * t) * sh;
        int   j0   = (int)fminf(fmaxf(floorf(src2), 0.0f), (float)(SDIM - 1));
        colKb[b*NT + t] = imin(j0, SDIM - KW);
    }
}

// ---------------------------------------------------------------------------
// Kernel 2: materialize banded interp coefficient tiles (16 x 48) with the
// sigmoid mask folded in. grid = (batch, tile, axis).
// ---------------------------------------------------------------------------
__global__ __launch_bounds__(256) void racnn_coef(
    const float* __restrict__ rowFr, const float* __restrict__ colFr,
    const float* __restrict__ mrow,  const float* __restrict__ mcol,
    const int*   __restrict__ rowI0, const int*   __restrict__ colI0,
    const int*   __restrict__ rowKb, const int*   __restrict__ colKb,
    float* __restrict__ rowCoef, float* __restrict__ colCoef)
{
    int b    = blockIdx.x;
    int t    = blockIdx.y;
    int axis = blockIdx.z;   // 0 = row (w), 1 = col (h)

    const int*   I0 = axis ? colI0 : rowI0;
    const float* Fr = axis ? colFr : rowFr;
    const float* Mk = axis ? mcol  : mrow;
    int          kb = (axis ? colKb : rowKb)[b*NT + t];
    float*      dst = (axis ? colCoef : rowCoef) + (size_t)(b*NT + t) * (16 * KW);

    for (int idx = threadIdx.x; idx < 16 * KW; idx += 256) {
        int r  = idx / KW;
        int k  = idx - r * KW;
        int gi = kb + k;
        int rg = t * 16 + r;

        int   i0 = I0[b*OUTN + rg];
        float fr = Fr[b*OUTN + rg];
        int   i1 = imin(i0 + 1, SDIM - 1);

        float cv = 0.0f;
        if (gi == i0) cv += 1.0f - fr;
        if (gi == i1) cv += fr;
        dst[idx] = cv * Mk[b*SDIM + gi];
    }
}

// ---------------------------------------------------------------------------
// Kernel 3: fused two-stage banded GEMM per 16x16 output tile, fp32 WMMA.
//   stage 1: tmp(16xKW) = Rtile(16xKW) * imgWin(KW x KW)
//   stage 2: out(16x16) = tmp(16xKW)  * CtileT(KW x 16)
// One wave32 per block, all 3 channels per block (coefficients hoisted to
// registers and reused). Image window stored COLUMN-major in LDS so each
// WMMA B operand is one aligned ds_load_b64; stride 52 makes the 32 lanes'
// 64-bit reads tile all 64 LDS banks exactly once (zero conflicts).
// ---------------------------------------------------------------------------
__global__ __launch_bounds__(32) void racnn_wmma(
    const float* __restrict__ img,
    const float* __restrict__ rowCoef, const float* __restrict__ colCoef,
    const int*   __restrict__ rowKb,   const int*   __restrict__ colKb,
    float* __restrict__ out)
{
    __shared__ float imgT[KW * IST];   // column-major: imgT[col*IST + row]
    __shared__ float tmpT[16 * TST];   // row-major:    tmpT[m*TST + k]

    int lane = threadIdx.x;
    int tr = blockIdx.x / NT;
    int ts = blockIdx.x - tr * NT;
    int b  = blockIdx.y;

    int kb = rowKb[b*NT + tr];   // image-row band start
    int jb = colKb[b*NT + ts];   // image-col band start

    int m  = lane & 15;     // N (or M) index within a lane half
    int hi = lane >> 4;     // 0 = lanes 0-15, 1 = lanes 16-31

    // Hoist channel-invariant coefficient tiles into registers.
    // A layout (16x4 f32): lane m holds K = 4kk + 2*hi + {0,1}  -> v2f pair.
    // B layout mirrored: lane holds (K = 4kk + 2*hi + {0,1}, N = m).
    const float* Rt = rowCoef + (size_t)(b*NT + tr) * (16 * KW) + m * KW + 2 * hi;
    const float* Ct = colCoef + (size_t)(b*NT + ts) * (16 * KW) + m * KW + 2 * hi;
    v2f areg[KCH], creg[KCH];
    #pragma unroll
    for (int kk = 0; kk < KCH; ++kk) {
        areg[kk] = *(const v2f*)(Rt + 4 * kk);
        creg[kk] = *(const v2f*)(Ct + 4 * kk);
    }

    for (int c = 0; c < CH; ++c) {
        const float* imgbc = img + (size_t)(b*CH + c) * SDIM * SDIM;

        // transpose-load the 48x48 window into LDS (coalesced global reads)
        for (int idx = lane; idx < KW * KW; idx += 32) {
            int rr = idx / KW;
            int cc = idx - rr * KW;
            imgT[cc * IST + rr] = imgbc[(size_t)(kb + rr) * SDIM + (jb + cc)];
        }
        __syncthreads();

        // ---- stage 1: three 16x16 N-tiles of tmp ----
        #pragma unroll
        for (int n = 0; n < 3; ++n) {
            v8f acc = {};
            #pragma unroll
            for (int kk = 0; kk < KCH; ++kk) {
                v2f bv = *(const v2f*)&imgT[(16*n + m) * IST + 4*kk + 2*hi];
                acc = __builtin_amdgcn_wmma_f32_16x16x4_f32(
                    false, areg[kk], false, bv, (short)0, acc, false, false);
            }
            // C layout -> row-major LDS (M = v + 8*hi, N = m)
            #pragma unroll
            for (int v = 0; v < 8; ++v)
                tmpT[(v + 8*hi) * TST + 16*n + m] = acc[v];
        }
        __syncthreads();

        // ---- stage 2: out(16x16) = tmp * C^T ----
        v8f acc = {};
        #pragma unroll
        for (int kk = 0; kk < KCH; ++kk) {
            v2f a = *(const v2f*)&tmpT[m * TST + 4*kk + 2*hi];
            acc = __builtin_amdgcn_wmma_f32_16x16x4_f32(
                false, a, false, creg[kk], (short)0, acc, false, false);
        }

        #pragma unroll
        for (int v = 0; v < 8; ++v) {
            int row = tr * 16 + v + 8*hi;
            int col = ts * 16 + m;
            out[((size_t)(b*CH + c) * OUTN + row) * OUTN + col] = acc[v];
        }
        __syncthreads();   // imgT/tmpT reused by next channel
    }
}

// ---------------------------------------------------------------------------
extern "C" void kernel_launch(void* const* d_in, const int* in_sizes, int n_in,
                              void* d_out, int out_size, void* d_ws, size_t ws_size,
                              hipStream_t stream)
{
    const float* images = (const float*)d_in[0];
    const float* locs   = (const float*)d_in[1];
    float*       out    = (float*)d_out;
    float*       ws     = (float*)d_ws;

    size_t off = 0;
    float* rowFr   = ws + off; off += (size_t)BATCH * OUTN;
    float* colFr   = ws + off; off += (size_t)BATCH * OUTN;
    float* mrow    = ws + off; off += (size_t)BATCH * SDIM;
    float* mcol    = ws + off; off += (size_t)BATCH * SDIM;
    float* rowCoef = ws + off; off += (size_t)BATCH * NT * 16 * KW;
    float* colCoef = ws + off; off += (size_t)BATCH * NT * 16 * KW;
    int* rowI0 = (int*)(ws + off); off += (size_t)BATCH * OUTN;
    int* colI0 = (int*)(ws + off); off += (size_t)BATCH * OUTN;
    int* rowKb = (int*)(ws + off); off += (size_t)BATCH * NT;
    int* colKb = (int*)(ws + off); off += (size_t)BATCH * NT;
    // total ~6 MB of workspace

    racnn_prep<<<BATCH, OUTN, 0, stream>>>(locs, rowFr, colFr, mrow, mcol,
                                           rowI0, colI0, rowKb, colKb);

    dim3 gc(BATCH, NT, 2);
    racnn_coef<<<gc, 256, 0, stream>>>(rowFr, colFr, mrow, mcol,
                                       rowI0, colI0, rowKb, colKb,
                                       rowCoef, colCoef);

    dim3 gm(NT * NT, BATCH);
    racnn_wmma<<<gm, 32, 0, stream>>>(images, rowCoef, colCoef,
                                      rowKb, colKb, out);
}